// MTH_18622978195602
// MI455X (gfx1250) — compile-verified
//
#include <hip/hip_runtime.h>
#include <cstdint>

// Multi-threshold spiking-neuron scan (T=4, K=8) for MI455X / gfx1250.
//
// Memory-bound streaming kernel: 128 MiB in + 128 MiB out => ~11.5 us floor
// at 23.3 TB/s. No contraction dim => WMMA inapplicable. Data path:
// global_load_async_to_lds_b128 (ASYNCcnt) -> ds_load_b128 -> registers,
// NT b128 stores out. Threshold select done exactly in the IEEE exponent
// domain (6 VALU/component instead of a 16-op cmp/cndmask chain), keeping
// VALU well under the HBM roofline.

#define T_STEPS 4
#define K_THRESH 8
#define BLOCK 256

typedef __attribute__((ext_vector_type(4))) float v4f;

// Exact first-crossed-threshold spike:
//   spike = thresh * 2^-k*,  k* = min k with mem >= 0.75*thresh*2^-k, else 0.
// For positive normals, scaling by 2^-k subtracts (k<<23) from the bit
// pattern, so k* = -min(0, (bits(mem)-bits(thr75_0)) >> 23). The single
// float compare against the smallest threshold handles mem<=0 / NaN / no-
// crossing, and guarantees k* <= 7 under the guard.
__device__ __forceinline__ float spike_of(float mem, float th, float thr75_7,
                                          int bits0) {
    const int e    = __builtin_bit_cast(int, mem) - bits0;
    const int m    = e >> 23;                // floor div by 2^23 (arith shift)
    const int expo = (m < 0) ? m : 0;        // v_min_i32 -> -k*
    const float s  = __builtin_ldexpf(th, expo);  // v_ldexp_f32, exact
    return (mem >= thr75_7) ? s : 0.0f;      // v_cmp + v_cndmask
}

__global__ __launch_bounds__(BLOCK) void snn_multithresh_kernel(
    const float* __restrict__ x,
    const float* __restrict__ thresh_ptr,
    float* __restrict__ out,
    int n_spatial)  // elements per timestep (B*C*H*W)
{
    // LDS staging: [T][BLOCK] float4 slots = 16 KiB per block.
    __shared__ __align__(16) float lds_tile[T_STEPS * BLOCK * 4];

    const int tid = threadIdx.x;
    const long long base = ((long long)blockIdx.x * BLOCK + tid) * 4;
    if (base >= (long long)n_spatial) return;

    // Low 32 bits of the generic pointer to a __shared__ object are the LDS
    // byte offset -- this is what the async-to-LDS VDST operand wants.
    const uint32_t lds_base =
        (uint32_t)(uintptr_t)(&lds_tile[0]) + (uint32_t)tid * 16u;

    // Kick off all T async global->LDS B128 copies (ASYNCcnt-tracked).
    // GVS addressing: saddr = x (SGPR pair), vaddr = 32-bit byte offset
    // (max offset = 2^27 bytes, fits signed i32).
#pragma unroll
    for (int t = 0; t < T_STEPS; ++t) {
        const uint32_t lds_addr = lds_base + (uint32_t)t * (BLOCK * 16u);
        const int goff = (int)((base + (long long)t * n_spatial) * 4);
        asm volatile("global_load_async_to_lds_b128 %0, %1, %2"
                     :: "v"(lds_addr), "v"(goff), "s"(x)
                     : "memory");
    }

    // Uniform threshold constants (SGPR-resident).
    const float th      = thresh_ptr[0];
    const float thr75_0 = 0.75f * th;                  // largest 0.75*threshold
    const float thr75_7 = __builtin_ldexpf(thr75_0, -(K_THRESH - 1));  // smallest
    const int   bits0   = __builtin_bit_cast(int, thr75_0);

    // Wait for all four async copies to land in LDS.
    asm volatile("s_wait_asynccnt 0x0" ::: "memory");

    v4f mem = (v4f)(0.0f);

#pragma unroll
    for (int t = 0; t < T_STEPS; ++t) {
        const v4f xv =
            *reinterpret_cast<const v4f*>(&lds_tile[(t * BLOCK + tid) * 4]);
        mem += xv;

        v4f spk;
        spk.x = spike_of(mem.x, th, thr75_7, bits0);
        spk.y = spike_of(mem.y, th, thr75_7, bits0);
        spk.z = spike_of(mem.z, th, thr75_7, bits0);
        spk.w = spike_of(mem.w, th, thr75_7, bits0);

        // Write-once output: non-temporal b128 store.
        v4f* outp = reinterpret_cast<v4f*>(out + base + (long long)t * n_spatial);
        __builtin_nontemporal_store(spk, outp);

        mem -= spk;
    }
}

extern "C" void kernel_launch(void* const* d_in, const int* in_sizes, int n_in,
                              void* d_out, int out_size, void* d_ws, size_t ws_size,
                              hipStream_t stream) {
    (void)n_in; (void)out_size; (void)d_ws; (void)ws_size;
    const float* x      = (const float*)d_in[0];
    const float* thresh = (const float*)d_in[1];
    float*       out    = (float*)d_out;

    const int n_total   = in_sizes[0];            // T*B*C*H*W = 33,554,432
    const int n_spatial = n_total / T_STEPS;      // 8,388,608 (divides exactly)
    const int n_vec     = n_spatial / 4;          // one float4 column per thread
    const int n_blocks  = (n_vec + BLOCK - 1) / BLOCK;  // 8192

    snn_multithresh_kernel<<<n_blocks, BLOCK, 0, stream>>>(x, thresh, out, n_spatial);
}